// WaveletScatteringTransform_31086973288769
// MI455X (gfx1250) — compile-verified
//
#include <hip/hip_runtime.h>
#include <math.h>

// Wavelet scattering (J=8,Q=8,T=65536,B=8) for gfx1250.
// Big FFTs: four-step 256x256, radix-16 stages as complex 16x16x16 matmuls on
// V_WMMA_F32_16X16X4_F32 (fp32).  Spectra are kept digit-permuted -> in-place.
// Global data: addrspace(1) ext-vector float2; LDS tiles: addrspace(3)
// (global_load/store_b64 + ds ops, no FLAT).  Next-line operands are
// software-prefetched (global_prefetch_b8) to hide HBM latency in the
// strided column phase while WMMA works on the current line.
// Workspace: ~509 MB (xf 4MB + U1f 268MB + scratch 235MB + G 1MB).

typedef __attribute__((ext_vector_type(2))) float v2f;   // complex / A,B operand
typedef __attribute__((ext_vector_type(8))) float v8f;   // C/D accumulator

typedef __attribute__((address_space(1))) v2f   gv2;     // global complex
typedef __attribute__((address_space(1))) float gfloat;  // global float
typedef __attribute__((address_space(3))) float lf;      // LDS float

#define PI_F 3.14159265358979f

__device__ inline v8f cwmma(v2f a, v2f b, v8f c) {
  // D = A(16x4) x B(4x16) + C(16x16), fp32
  return __builtin_amdgcn_wmma_f32_16x16x4_f32(false, a, false, b, (short)0, c,
                                               false, false);
}

// Lane-invariant twiddle operands, computed once per big-FFT direction.
// Fr/Fi[j]: DFT-16 factors for K-pair {4j+2kk, 4j+2kk+1} at row/col m.
// (Stage-1 A operand == stage-2 B operand, same values.)
// tc/ts[v]: internal 256-twiddle for C element (k2=v+8kk, n1=m).
struct Tw { v2f Fr[4], Fi[4], FiN[4]; float tc[8], ts[8]; };

__device__ inline void make_tw(Tw& t, float sign) {
  const int lane = threadIdx.x & 31;
  const int m = lane & 15, kk = lane >> 4;
  const float c16  = sign * (2.0f * PI_F / 16.0f);
  const float c256 = sign * (2.0f * PI_F / 256.0f);
#pragma unroll
  for (int j = 0; j < 4; ++j) {
    const int ka = 4*j + 2*kk, kb = ka + 1;
    float sa, ca, sb, cb;
    __sincosf(c16 * (float)((m*ka) & 15), &sa, &ca);
    __sincosf(c16 * (float)((m*kb) & 15), &sb, &cb);
    t.Fr[j]  = (v2f){ca, cb};
    t.Fi[j]  = (v2f){sa, sb};
    t.FiN[j] = (v2f){-sa, -sb};
  }
#pragma unroll
  for (int v = 0; v < 8; ++v) {
    const int k2 = v + 8*kk;
    __sincosf(c256 * (float)(k2 * m), &t.ts[v], &t.tc[v]);
  }
}

// ---------------------------------------------------------------------------
// One wave32: 256-point DFT (sign=-1) / unnormalized IDFT (sign=+1) on
// buf[i*stride], i=0..255, in place.  Y = F16 * Xm * F16^T with the internal
// 256-twiddle in between; natural-order output.  If bigTw, also applies the
// four-step twiddle exp(sign*2pi*i*outIdx*lineId/65536).
// pre: base of the NEXT line this wave will transform (nullptr = none);
// its operand addresses are prefetched to overlap memory with WMMA.
// tR/tI: per-wave 16x17 LDS tiles (C-layout -> A-layout relayout).
// Contains 2 x __syncthreads(): every wave of the block must call uniformly.
// ---------------------------------------------------------------------------
__device__ void fft256_wave(gv2* buf, const gv2* pre, int stride, float sign,
                            int lineId, bool bigTw,
                            lf (*tR)[17], lf (*tI)[17], const Tw& t) {
  const int lane = threadIdx.x & 31;
  const int m  = lane & 15;
  const int kk = lane >> 4;
  const float cBig = sign * (2.0f * PI_F / 65536.0f);

  // ---- stage 1: M1 = F16 x Xm ; Xm[n2][n1] = x[n1 + 16*n2]
  v8f Cr = {0.f,0.f,0.f,0.f,0.f,0.f,0.f,0.f};
  v8f Ci = {0.f,0.f,0.f,0.f,0.f,0.f,0.f,0.f};
#pragma unroll
  for (int j = 0; j < 4; ++j) {
    const int ka = 4*j + 2*kk, kb = ka + 1;          // contraction n2
    v2f xa = buf[(m + 16*ka) * stride];              // B operand direct load
    v2f xb = buf[(m + 16*kb) * stride];
    if (pre) {                                       // cover next line (8/lane)
      __builtin_prefetch((const v2f*)&pre[(m + 16*ka) * stride], 0, 1);
      __builtin_prefetch((const v2f*)&pre[(m + 16*kb) * stride], 0, 1);
    }
    v2f Br = {xa.x, xb.x}, Bi = {xa.y, xb.y};
    Cr = cwmma(t.Fr[j],  Br, Cr);  Cr = cwmma(t.FiN[j], Bi, Cr);
    Ci = cwmma(t.Fr[j],  Bi, Ci);  Ci = cwmma(t.Fi[j],  Br, Ci);
  }

  // ---- internal 256-twiddle, then C-layout -> A-layout via LDS
  __syncthreads();
#pragma unroll
  for (int v = 0; v < 8; ++v) {
    const int k2 = v + 8*kk;                         // C row (M), col N = m
    tR[k2][m] = Cr[v]*t.tc[v] - Ci[v]*t.ts[v];
    tI[k2][m] = Cr[v]*t.ts[v] + Ci[v]*t.tc[v];
  }
  __syncthreads();

  // ---- stage 2: C2 = T x F16^T  (A = T from LDS, B = F16 factors)
  v8f Er = {0.f,0.f,0.f,0.f,0.f,0.f,0.f,0.f};
  v8f Ei = {0.f,0.f,0.f,0.f,0.f,0.f,0.f,0.f};
#pragma unroll
  for (int j = 0; j < 4; ++j) {
    const int ka = 4*j + 2*kk, kb = ka + 1;          // contraction n1
    v2f Ar  = {tR[m][ka], tR[m][kb]};
    v2f Ai  = {tI[m][ka], tI[m][kb]};
    v2f AiN = {-Ai.x, -Ai.y};
    Er = cwmma(Ar,  t.Fr[j], Er);  Er = cwmma(AiN, t.Fi[j], Er);
    Ei = cwmma(Ar,  t.Fi[j], Ei);  Ei = cwmma(Ai,  t.Fr[j], Ei);
  }

  // ---- store: element [k2][k1] -> Y[16*k1 + k2]; optional big twiddle
  const float base = cBig * (float)lineId;
#pragma unroll
  for (int v = 0; v < 8; ++v) {
    const int k2 = v + 8*kk, k1 = m;
    const int idx = 16*k1 + k2;
    float er = Er[v], ei = Ei[v];
    if (bigTw) {
      float s, c; __sincosf(base * (float)idx, &s, &c);
      float tmp = er*c - ei*s; ei = er*s + ei*c; er = tmp;
    }
    buf[idx * stride] = (v2f){er, ei};
  }
}

// ---------------------------------------------------------------------------
// In-place 65536-pt transform on `line`, one 256-thread block (8 waves).
// Digit-permuted frequency order: slot 256*k1+k2 <-> true freq k1+256*k2.
// sign=-1: natural time -> permuted freq.  sign=+1: permuted freq -> natural
// time (unnormalized; caller applies 1/65536).
// ---------------------------------------------------------------------------
__device__ void bigfft_inplace(gv2* line, float sign,
                               lf (*tR)[17], lf (*tI)[17]) {
  const int wave = threadIdx.x >> 5;
  Tw tw; make_tw(tw, sign);
  if (sign < 0.0f) {
    for (int c = wave; c < 256; c += 8)
      fft256_wave(line + c, (c + 8 < 256) ? line + c + 8 : nullptr,
                  256, sign, c, true, tR, tI, tw);
    __threadfence(); __syncthreads();
    for (int r = wave; r < 256; r += 8)
      fft256_wave(line + (size_t)256*r,
                  (r + 8 < 256) ? line + (size_t)256*(r+8) : nullptr,
                  1, sign, r, false, tR, tI, tw);
  } else {
    for (int r = wave; r < 256; r += 8)
      fft256_wave(line + (size_t)256*r,
                  (r + 8 < 256) ? line + (size_t)256*(r+8) : nullptr,
                  1, sign, r, true, tR, tI, tw);
    __threadfence(); __syncthreads();
    for (int c = wave; c < 256; c += 8)
      fft256_wave(line + c, (c + 8 < 256) ? line + c + 8 : nullptr,
                  256, sign, c, false, tR, tI, tw);
  }
  __threadfence(); __syncthreads();
}

// Fold spectrum*phi into 256 aliased bins (includes the 1/N ifft scale).
// Thread m reads the contiguous row vf[256*m .. 256*m+255] (permuted layout).
__device__ void lowpass_fold(const gv2* vf, gv2* G) {
  const int mrow = threadIdx.x;              // 256 threads
  const float sigp = 0.35f / 256.0f;         // phi sigma (normalized freq)
  float sr = 0.f, si = 0.f;
  const gv2* row = vf + (size_t)256 * mrow;
  for (int j = 0; j < 256; ++j) {
    int kt = mrow + (j << 8);                // true frequency bin
    float fs = (float)(kt < 32768 ? kt : kt - 65536) * (1.0f / 65536.0f);
    float d = fs / sigp;
    float g = __expf(-0.5f * d * d);
    v2f v = row[j];
    sr += v.x * g; si += v.y * g;
  }
  G[mrow] = (v2f){sr * (1.0f/65536.0f), si * (1.0f/65536.0f)};
}

__device__ void write_channel(const gv2* G, gfloat* out, int b, int ch) {
  const int tpos = threadIdx.x;
  v2f gv = G[tpos];
  float v = gv.x;                            // real(ifft(..)) at stride 256
  float mag = sqrtf(v*v + 1e-8f);
  size_t o = ((size_t)(b*289 + ch))*256 + tpos;
  out[o] = logf(mag + 1e-8f);                // "real" block
  out[o + (size_t)8*289*256] = 0.0f;         // "imag" block = 0
}

// ------------------------------- kernels -----------------------------------

__global__ __launch_bounds__(256) void k_fft_x(const float* __restrict__ x_,
                                               v2f* __restrict__ xf_) {
  __shared__ float sR[8][16][17], sI[8][16][17];
  lf (*tR)[17] = (lf (*)[17])(sR[threadIdx.x >> 5]);
  lf (*tI)[17] = (lf (*)[17])(sI[threadIdx.x >> 5]);
  const int b = blockIdx.x, tid = threadIdx.x;
  gv2* line = (gv2*)(xf_ + (size_t)b * 65536);
  const gfloat* xr = (const gfloat*)(x_ + (size_t)b * 65536);
  for (int i = tid; i < 65536; i += 256) line[i] = (v2f){xr[i], 0.0f};
  __threadfence(); __syncthreads();
  bigfft_inplace(line, -1.0f, tR, tI);       // -> permuted spectrum
}

__global__ __launch_bounds__(256) void k_order1(const v2f* __restrict__ xf_,
                                                v2f* __restrict__ u1f_) {
  __shared__ float sR[8][16][17], sI[8][16][17];
  lf (*tR)[17] = (lf (*)[17])(sR[threadIdx.x >> 5]);
  lf (*tI)[17] = (lf (*)[17])(sI[threadIdx.x >> 5]);
  const int id = blockIdx.x, tid = threadIdx.x;
  const int b = id >> 6, l1 = id & 63;
  gv2* line = (gv2*)(u1f_ + (size_t)id * 65536);
  const gv2* xb = (const gv2*)(xf_ + (size_t)b * 65536);
  const float xi = 0.35f * exp2f(-(float)l1 / 8.0f);  // xis1[l1]
  const float inv_sig = 8.0f / xi;                    // 1/sig1
  for (int p = tid; p < 65536; p += 256) {            // Yf = xf * psi1
    int k1 = p >> 8, k2 = p & 255, kt = k1 + (k2 << 8);
    float fs = (float)(kt < 32768 ? kt : kt - 65536) * (1.0f / 65536.0f);
    float d = (fs - xi) * inv_sig;
    float g = __expf(-0.5f * d * d);
    v2f v = xb[p];
    line[p] = (v2f){v.x * g, v.y * g};
  }
  __threadfence(); __syncthreads();
  bigfft_inplace(line, +1.0f, tR, tI);                // -> time (unnormalized)
  for (int p = tid; p < 65536; p += 256) {            // U1 = |.|/N
    v2f v = line[p];
    line[p] = (v2f){sqrtf(v.x*v.x + v.y*v.y) * (1.0f/65536.0f), 0.0f};
  }
  __threadfence(); __syncthreads();
  bigfft_inplace(line, -1.0f, tR, tI);                // -> U1f (permuted)
}

__global__ __launch_bounds__(256) void k_lowpass1(const v2f* __restrict__ xf_,
                                                  const v2f* __restrict__ u1f_,
                                                  v2f* __restrict__ gbuf_,
                                                  float* __restrict__ out_) {
  __shared__ float sR[8][16][17], sI[8][16][17];
  lf (*tR)[17] = (lf (*)[17])(sR[threadIdx.x >> 5]);
  lf (*tI)[17] = (lf (*)[17])(sI[threadIdx.x >> 5]);
  const int id = blockIdx.x;                 // 8*65 blocks
  const int b = id / 65, ch = id % 65;       // ch 0 = S0, ch 1..64 = S1
  const gv2* vf = (ch == 0) ? (const gv2*)(xf_ + (size_t)b * 65536)
                            : (const gv2*)(u1f_ + (size_t)(b*64 + ch - 1) * 65536);
  gv2* G = (gv2*)(gbuf_ + (size_t)id * 256);
  lowpass_fold(vf, G);
  __threadfence(); __syncthreads();
  Tw tw; make_tw(tw, +1.0f);
  // 256-pt IDFT of folded bins (all waves redundantly, identical values)
  fft256_wave(G, nullptr, 1, +1.0f, 0, false, tR, tI, tw);
  __threadfence(); __syncthreads();
  write_channel(G, (gfloat*)out_, b, ch);
}

__global__ __launch_bounds__(256) void k_order2(const v2f* __restrict__ u1f_,
                                                v2f* __restrict__ sc_,
                                                v2f* __restrict__ gbuf_,
                                                float* __restrict__ out_,
                                                int j2) {
  __shared__ float sR[8][16][17], sI[8][16][17];
  lf (*tR)[17] = (lf (*)[17])(sR[threadIdx.x >> 5]);
  lf (*tI)[17] = (lf (*)[17])(sI[threadIdx.x >> 5]);
  const int cnt = 8 * j2;                    // # of l1 with j1 < j2
  const int id = blockIdx.x, tid = threadIdx.x;
  const int b = id / cnt, l1 = id % cnt;
  const gv2* vf = (const gv2*)(u1f_ + (size_t)(b*64 + l1) * 65536);
  gv2* line = (gv2*)(sc_ + (size_t)id * 65536);
  const float xi = 0.35f * exp2f(-(float)j2);        // xis2[j2]
  const float inv_sig = 1.0f / (0.6f * xi);
  for (int p = tid; p < 65536; p += 256) {           // U1f * psi2
    int k1 = p >> 8, k2 = p & 255, kt = k1 + (k2 << 8);
    float fs = (float)(kt < 32768 ? kt : kt - 65536) * (1.0f / 65536.0f);
    float d = (fs - xi) * inv_sig;
    float g = __expf(-0.5f * d * d);
    v2f v = vf[p];
    line[p] = (v2f){v.x * g, v.y * g};
  }
  __threadfence(); __syncthreads();
  bigfft_inplace(line, +1.0f, tR, tI);               // -> time
  for (int p = tid; p < 65536; p += 256) {           // U2 = |.|/N
    v2f v = line[p];
    line[p] = (v2f){sqrtf(v.x*v.x + v.y*v.y) * (1.0f/65536.0f), 0.0f};
  }
  __threadfence(); __syncthreads();
  bigfft_inplace(line, -1.0f, tR, tI);               // -> U2f (permuted)
  gv2* G = (gv2*)(gbuf_ + (size_t)id * 256);
  lowpass_fold(line, G);
  __threadfence(); __syncthreads();
  Tw tw; make_tw(tw, +1.0f);
  fft256_wave(G, nullptr, 1, +1.0f, 0, false, tR, tI, tw);
  __threadfence(); __syncthreads();
  const int ch = 65 + 4*j2*(j2 - 1) + l1;            // channel layout offset
  write_channel(G, (gfloat*)out_, b, ch);
}

// ------------------------------ launcher -----------------------------------

extern "C" void kernel_launch(void* const* d_in, const int* in_sizes, int n_in,
                              void* d_out, int out_size, void* d_ws,
                              size_t ws_size, hipStream_t stream) {
  const float* x = (const float*)d_in[0];            // [8][65536] f32
  float* out = (float*)d_out;                        // real||imag, f32
  v2f* ws2  = (v2f*)d_ws;
  v2f* xf   = ws2;                                   //   8*65536 c64 =   4 MB
  v2f* u1f  = xf  + (size_t)8   * 65536;             // 512*65536 c64 = 268 MB
  v2f* sc   = u1f + (size_t)512 * 65536;             // 448*65536 c64 = 235 MB
  v2f* gbuf = sc  + (size_t)448 * 65536;             // 520*256   c64 =   1 MB

  k_fft_x   <<<8,   256, 0, stream>>>(x, xf);
  k_order1  <<<512, 256, 0, stream>>>(xf, u1f);
  k_lowpass1<<<520, 256, 0, stream>>>(xf, u1f, gbuf, out);
  for (int j2 = 1; j2 < 8; ++j2)                     // S2: 8*8*j2 lines each
    k_order2<<<8 * 8 * j2, 256, 0, stream>>>(u1f, sc, gbuf, out, j2);

  (void)in_sizes; (void)n_in; (void)out_size; (void)ws_size;
}